// FreqEnhanceBlock_42116449304634
// MI455X (gfx1250) — compile-verified
//
#include <hip/hip_runtime.h>
#include <math.h>

// ---------------------------------------------------------------------------
// FreqEnhanceBlock, algebraically reduced:
//   out[b,c,2d+p,2h+q,2w+r] = w * x[...] + (1-w) * mean8(2x2x2 block)
//   w = sigmoid( MLP_g(gap)[b,c] + MLP_l(2*sqrt(2)*x[b,c,2d,2h,2w])[b,c,d,h,w] )
//   gap[b,c] = 2*sqrt(2) * mean_{d,h,w}( x[b,c,2d,2h,2w] )
// Shapes: B=2, C=128, D=H=W=64, mid=8.  Pure HBM-bound: ~512 MB traffic.
// ---------------------------------------------------------------------------

typedef __attribute__((ext_vector_type(2))) float v2f;
typedef __attribute__((ext_vector_type(8))) float v8f;

#define C_CH   128
#define MID    8
#define S3     262144   // 64*64*64
#define PLANE  4096     // 64*64
#define ROW    64
#define SQ8    2.8284271247461903f  // 2*sqrt(2)

// ---------------------------------------------------------------------------
// Kernel A: gap[b,c] = 2*sqrt(2)/32768 * sum over 32^3 of x[b,c,2d,2h,2w]
// ---------------------------------------------------------------------------
__global__ __launch_bounds__(256)
void gap_reduce_kernel(const float* __restrict__ x, float* __restrict__ gap) {
    const int bc = blockIdx.x;                 // 0..255 = b*128 + c
    const float* xb = x + (size_t)bc * S3;
    float s = 0.f;
    for (int i = threadIdx.x; i < 32768; i += 256) {
        const int w = i & 31, h = (i >> 5) & 31, d = i >> 10;
        s += xb[d * (2 * PLANE) + h * (2 * ROW) + 2 * w];
    }
    __shared__ float red[256];
    red[threadIdx.x] = s;
    __syncthreads();
    for (int off = 128; off > 0; off >>= 1) {
        if (threadIdx.x < off) red[threadIdx.x] += red[threadIdx.x + off];
        __syncthreads();
    }
    if (threadIdx.x == 0) gap[bc] = red[0] * (SQ8 / 32768.f);
}

// ---------------------------------------------------------------------------
// Kernel B: g_out[b,c] = MLP_g(gap[b,:])[c]   (tiny: 2*128 outputs, 1 block)
// ---------------------------------------------------------------------------
__global__ __launch_bounds__(256)
void gmlp_kernel(const float* __restrict__ gap,
                 const float* __restrict__ gw1, const float* __restrict__ gb1,
                 const float* __restrict__ gw2, const float* __restrict__ gb2,
                 float* __restrict__ g_out) {
    __shared__ float g[2][C_CH];
    const int tid = threadIdx.x;               // 256 threads
    g[tid >> 7][tid & 127] = gap[tid];
    __syncthreads();
    const int b = tid >> 7, c = tid & 127;
    float hbuf[MID];
#pragma unroll
    for (int m = 0; m < MID; ++m) {
        float a = gb1[m];
        for (int k = 0; k < C_CH; ++k) a = fmaf(gw1[m * C_CH + k], g[b][k], a);
        hbuf[m] = fmaxf(a, 0.f);
    }
    float o = gb2[c];
#pragma unroll
    for (int m = 0; m < MID; ++m) o = fmaf(gw2[c * MID + m], hbuf[m], o);
    g_out[tid] = o;
}

// ---------------------------------------------------------------------------
// Kernel C: fused main pass. One workgroup per (b, coarse-d, coarse-h) row:
//   32 coarse-w positions x 128 channels.
//   Phase 1: stage t = 2*sqrt(2)*x_even into LDS (B128 loads).
//   MLP (waves 0-1): V_WMMA_F32_16X16X4_F32  layer1 (K=128) + layer2 (N=128).
//   Phase 2: stream 8 corners with B128 loads/stores, blend
//            out = w*x + (1-w)*mean8; each thread owns 2 coarse blocks.
// ---------------------------------------------------------------------------
__global__ __launch_bounds__(256)
void freq_main_kernel(const float* __restrict__ x,
                      const float* __restrict__ lw1, const float* __restrict__ lb1,
                      const float* __restrict__ lw2, const float* __restrict__ lb2,
                      const float* __restrict__ g_out,
                      float* __restrict__ out) {
    __shared__ __attribute__((aligned(16))) float t_lds[32 * 130]; // [pos][c]
    __shared__ __attribute__((aligned(16))) float w1p [16 * 130];  // [n][k], n pad->16
    __shared__ __attribute__((aligned(16))) float w2p [C_CH * 9];  // [c][k]
    __shared__ __attribute__((aligned(16))) float h_lds[32 * 17];  // [pos][m]
    __shared__ __attribute__((aligned(16))) float w_lds[C_CH * 34];// [c][pos], pad 34
    __shared__ float gb_lds[C_CH];      // lb2[c] + g_out[b,c]
    __shared__ float b1_lds[16];        // lb1 padded with zeros

    const int tid = threadIdx.x;
    const int blk = blockIdx.x;                 // b*1024 + d*32 + h
    const int b = blk >> 10;
    const int d = (blk >> 5) & 31;
    const int h = blk & 31;

    const float* xb = x   + (size_t)b * C_CH * S3 + d * (2 * PLANE) + h * (2 * ROW);
    float*       ob = out + (size_t)b * C_CH * S3 + d * (2 * PLANE) + h * (2 * ROW);

    // ---- stage weights ----
    for (int i = tid; i < 16 * C_CH; i += 256) {
        const int n = i >> 7, k = i & 127;
        w1p[n * 130 + k] = (n < MID) ? lw1[n * C_CH + k] : 0.f;
    }
    for (int i = tid; i < C_CH * MID; i += 256) {
        const int c = i >> 3, k = i & 7;
        w2p[c * 9 + k] = lw2[i];
    }
    if (tid < C_CH) gb_lds[tid] = lb2[tid] + g_out[b * C_CH + tid];
    if (tid < 16)   b1_lds[tid] = (tid < MID) ? lb1[tid] : 0.f;

    // ---- stage t = 2*sqrt(2) * x[b,c,2d,2h,2w]  (B128 reads, keep evens) ----
    {
        const int wp = tid & 15;                // covers wc = 2wp, 2wp+1
        const int c0 = tid >> 4;                // 0..15
        for (int cg = 0; cg < 8; ++cg) {
            const int c = cg * 16 + c0;
            const float4 v = *(const float4*)(xb + (size_t)c * S3 + 4 * wp);
            t_lds[(2 * wp) * 130 + c]     = SQ8 * v.x;
            t_lds[(2 * wp + 1) * 130 + c] = SQ8 * v.z;
        }
    }
    __syncthreads();

    // ---- MLP via WMMA (waves 0 and 1 handle 16 positions each) ----
    const int wave = tid >> 5;
    const int lane = tid & 31;
    const int rown = lane & 15;                 // M row / N col within tile
    const int hi   = lane >> 4;                 // selects K pair {0,1} vs {2,3}
    if (wave < 2) {
        const int pos = wave * 16 + rown;
        // layer 1: acc[pos][n] = sum_k t[pos][k] * lw1[n][k]
        v8f acc = {0.f, 0.f, 0.f, 0.f, 0.f, 0.f, 0.f, 0.f};
        for (int k0 = 0; k0 < C_CH; k0 += 4) {
            const int k = k0 + hi * 2;
            v2f a;  a.x  = t_lds[pos * 130 + k];  a.y  = t_lds[pos * 130 + k + 1];
            v2f bb; bb.x = w1p [rown * 130 + k];  bb.y = w1p [rown * 130 + k + 1];
            acc = __builtin_amdgcn_wmma_f32_16x16x4_f32(
                      false, a, false, bb, (short)0, acc, false, false);
        }
        // D layout: VGPR j -> row = j + hi*8 (within the wave's 16-row tile)
        const float bias = b1_lds[rown];
#pragma unroll
        for (int j = 0; j < 8; ++j) {
            const float hv = fmaxf(acc[j] + bias, 0.f);
            h_lds[(wave * 16 + j + hi * 8) * 17 + rown] = hv;
        }
    }
    __syncthreads();
    if (wave < 2) {
        const int pos = wave * 16 + rown;
        // layer 2: out[pos][c] = sum_m h[pos][m] * lw2[c][m] ; c in 8 tiles of 16
        for (int tile = 0; tile < 8; ++tile) {
            v8f acc = {0.f, 0.f, 0.f, 0.f, 0.f, 0.f, 0.f, 0.f};
            const int c = tile * 16 + rown;
#pragma unroll
            for (int k0 = 0; k0 < MID; k0 += 4) {
                const int k = k0 + hi * 2;
                v2f a;  a.x  = h_lds[pos * 17 + k];  a.y  = h_lds[pos * 17 + k + 1];
                v2f bb; bb.x = w2p [c * 9 + k];      bb.y = w2p [c * 9 + k + 1];
                acc = __builtin_amdgcn_wmma_f32_16x16x4_f32(
                          false, a, false, bb, (short)0, acc, false, false);
            }
            const float gb = gb_lds[c];
#pragma unroll
            for (int j = 0; j < 8; ++j) {
                const float pre = acc[j] + gb;
                const float wv  = 1.f / (1.f + __expf(-pre));   // sigmoid
                w_lds[c * 34 + (wave * 16 + j + hi * 8)] = wv;
            }
        }
    }
    __syncthreads();

    // ---- phase 2: B128 streaming; each thread blends 2 coarse blocks ----
    {
        const int wp = tid & 15;                // wc = 2wp (elems .x.y), 2wp+1 (.z.w)
        const int c0 = tid >> 4;                // 0..15
        for (int cg = 0; cg < 8; ++cg) {
            const int c = cg * 16 + c0;
            const float* r = xb + (size_t)c * S3;
            float*       o = ob + (size_t)c * S3;
            if (cg + 1 < 8) {  // hint next channel-group (global_prefetch_b8)
                __builtin_prefetch(xb + (size_t)(c + 16) * S3 + 4 * wp, 0, 1);
            }
            const float4 a00 = *(const float4*)(r + 4 * wp);
            const float4 a01 = *(const float4*)(r + ROW + 4 * wp);
            const float4 a10 = *(const float4*)(r + PLANE + 4 * wp);
            const float4 a11 = *(const float4*)(r + PLANE + ROW + 4 * wp);
            const float m0 = (a00.x + a00.y + a01.x + a01.y +
                              a10.x + a10.y + a11.x + a11.y) * 0.125f;
            const float m1 = (a00.z + a00.w + a01.z + a01.w +
                              a10.z + a10.w + a11.z + a11.w) * 0.125f;
            const float2 wv = *(const float2*)&w_lds[c * 34 + 2 * wp];
            const float om0 = (1.f - wv.x) * m0;
            const float om1 = (1.f - wv.y) * m1;
            const float4 o00 = make_float4(fmaf(wv.x, a00.x, om0), fmaf(wv.x, a00.y, om0),
                                           fmaf(wv.y, a00.z, om1), fmaf(wv.y, a00.w, om1));
            const float4 o01 = make_float4(fmaf(wv.x, a01.x, om0), fmaf(wv.x, a01.y, om0),
                                           fmaf(wv.y, a01.z, om1), fmaf(wv.y, a01.w, om1));
            const float4 o10 = make_float4(fmaf(wv.x, a10.x, om0), fmaf(wv.x, a10.y, om0),
                                           fmaf(wv.y, a10.z, om1), fmaf(wv.y, a10.w, om1));
            const float4 o11 = make_float4(fmaf(wv.x, a11.x, om0), fmaf(wv.x, a11.y, om0),
                                           fmaf(wv.y, a11.z, om1), fmaf(wv.y, a11.w, om1));
            *(float4*)(o + 4 * wp)               = o00;
            *(float4*)(o + ROW + 4 * wp)         = o01;
            *(float4*)(o + PLANE + 4 * wp)       = o10;
            *(float4*)(o + PLANE + ROW + 4 * wp) = o11;
        }
    }
}

// ---------------------------------------------------------------------------
// Launch: inputs (dict order): x, gw1, gb1, gw2, gb2, lw1, lb1, lw2, lb2
// ---------------------------------------------------------------------------
extern "C" void kernel_launch(void* const* d_in, const int* in_sizes, int n_in,
                              void* d_out, int out_size, void* d_ws, size_t ws_size,
                              hipStream_t stream) {
    const float* x   = (const float*)d_in[0];
    const float* gw1 = (const float*)d_in[1];
    const float* gb1 = (const float*)d_in[2];
    const float* gw2 = (const float*)d_in[3];
    const float* gb2 = (const float*)d_in[4];
    const float* lw1 = (const float*)d_in[5];
    const float* lb1 = (const float*)d_in[6];
    const float* lw2 = (const float*)d_in[7];
    const float* lb2 = (const float*)d_in[8];
    float* out = (float*)d_out;

    float* gap   = (float*)d_ws;        // 256 floats
    float* g_out = gap + 256;           // 256 floats

    gap_reduce_kernel<<<256, 256, 0, stream>>>(x, gap);
    gmlp_kernel<<<1, 256, 0, stream>>>(gap, gw1, gb1, gw2, gb2, g_out);
    freq_main_kernel<<<2048, 256, 0, stream>>>(x, lw1, lb1, lw2, lb2, g_out, out);
}